// Transformer_66529043415377
// MI455X (gfx1250) — compile-verified
//
#include <hip/hip_runtime.h>

#define TPB 256
#define PPT 4   // pixels per thread

typedef float v4f __attribute__((ext_vector_type(4)));
typedef int   v4i __attribute__((ext_vector_type(4)));

#if defined(__has_builtin)
#if __has_builtin(__builtin_amdgcn_global_load_async_to_lds_b128)
#define USE_ASYNC_LDS 1
typedef __attribute__((address_space(1))) v4i* async_gptr_t;  // global v4i*
typedef __attribute__((address_space(3))) v4i* async_lptr_t;  // LDS    v4i*
#endif
#endif

// flow: (16, 2, 512, 512) f32 ; pqf: (16, 3, 512, 512) f32 viewed raw as [N,3]
// out : raw [N,3] -> d_out[i*3+c]
__global__ __launch_bounds__(TPB) void Transformer_warp_kernel(
    const float* __restrict__ flow, const float* __restrict__ pqf,
    float* __restrict__ out) {
  constexpr int HW = 512 * 512;  // 2^18
  // Staged gather footprint for this block's batch b:
  // record (y,x) -> tile[(y*512+x)*3 + c], y in {0,1,2}, x in {0..511}  (18 KB)
  __shared__ float tile[3 * 512 * 3];

  const int tid = threadIdx.x;
  const int blockPix0 = blockIdx.x * (TPB * PPT);
  const int b = blockPix0 >> 18;  // block-uniform (1024 divides 2^18)

  // ---- stage 3 clamp-rows of batch b into LDS ----
  // row y: 1536 contiguous floats at pqf + b*1536 + y*786432  (6144 B each)
#ifdef USE_ASYNC_LDS
  {
#pragma unroll
    for (int y = 0; y < 3; ++y) {
      const float* src = pqf + b * 1536 + y * 786432;
      for (int v = tid; v < 384; v += TPB) {
        __builtin_amdgcn_global_load_async_to_lds_b128(
            (async_gptr_t)(src + v * 4),
            (async_lptr_t)(&tile[y * 1536 + v * 4]),
            0, 0);
      }
    }
  }
#if __has_builtin(__builtin_amdgcn_s_wait_asynccnt)
  __builtin_amdgcn_s_wait_asynccnt(0);
#else
  asm volatile("s_wait_asynccnt 0" ::: "memory");
#endif
  __syncthreads();
#else
  {
#pragma unroll
    for (int y = 0; y < 3; ++y) {
      const v4f* src = (const v4f*)(pqf + b * 1536 + y * 786432);
      v4f* dst = (v4f*)&tile[y * 1536];
      for (int v = tid; v < 384; v += TPB) dst[v] = src[v];
    }
  }
  __syncthreads();
#endif

  // ---- per-thread: 4 consecutive pixels (same h row) ----
  const int t0 = blockPix0 + tid * PPT;
  const int hw0 = t0 & (HW - 1);
  const int h = hw0 >> 9;
  const int w0 = hw0 & 511;

  const float* fbase = flow + b * (2 * HW);
  const v4f dx4 = __builtin_nontemporal_load((const v4f*)(fbase + hw0));
  const v4f dy4 = __builtin_nontemporal_load((const v4f*)(fbase + HW + hw0));

  float res[12];
#pragma unroll
  for (int j = 0; j < 4; ++j) {
    const float x = (float)(w0 + j) + dx4[j] * 64.0f;
    const float y = (float)h + dy4[j] * 64.0f;
    float x0f = floorf(x), y0f = floorf(y);
    float x1f = x0f + 1.0f, y1f = y0f + 1.0f;
    // faithful quirky clamps: x to [0, H-1]=511 ; y to [0, C-1]=2
    x0f = fminf(fmaxf(x0f, 0.0f), 511.0f);
    x1f = fminf(fmaxf(x1f, 0.0f), 511.0f);
    y0f = fminf(fmaxf(y0f, 0.0f), 2.0f);
    y1f = fminf(fmaxf(y1f, 0.0f), 2.0f);
    const float wa = (x1f - x) * (y1f - y);
    const float wb = (x1f - x) * (y - y0f);
    const float wc = (x - x0f) * (y1f - y);
    const float wd = (x - x0f) * (y - y0f);
    const int x0 = (int)x0f, x1 = (int)x1f, y0 = (int)y0f, y1 = (int)y1f;
    const int ra = (y0 * 512 + x0) * 3;
    const int rb = (y1 * 512 + x0) * 3;
    const int rc = (y0 * 512 + x1) * 3;
    const int rd = (y1 * 512 + x1) * 3;
#pragma unroll
    for (int c = 0; c < 3; ++c) {
      res[j * 3 + c] = wa * tile[ra + c] + wb * tile[rb + c] +
                       wc * tile[rc + c] + wd * tile[rd + c];
    }
  }

  // coalesced 48B store per thread, 16B aligned (t0 % 4 == 0)
  v4f* o = (v4f*)(out + (size_t)t0 * 3);
  v4f o0; o0[0] = res[0]; o0[1] = res[1]; o0[2] = res[2];  o0[3] = res[3];
  v4f o1; o1[0] = res[4]; o1[1] = res[5]; o1[2] = res[6];  o1[3] = res[7];
  v4f o2; o2[0] = res[8]; o2[1] = res[9]; o2[2] = res[10]; o2[3] = res[11];
  __builtin_nontemporal_store(o0, o + 0);
  __builtin_nontemporal_store(o1, o + 1);
  __builtin_nontemporal_store(o2, o + 2);
}

extern "C" void kernel_launch(void* const* d_in, const int* in_sizes, int n_in,
                              void* d_out, int out_size, void* d_ws, size_t ws_size,
                              hipStream_t stream) {
  const float* flow = (const float*)d_in[0];  // (16,2,512,512) f32
  const float* pqf  = (const float*)d_in[1];  // (16,3,512,512) f32
  float* out = (float*)d_out;                 // (16,3,512,512) f32 (raw [N,3])
  (void)in_sizes; (void)n_in; (void)out_size; (void)d_ws; (void)ws_size;

  constexpr int N = 16 * 512 * 512;
  const dim3 grid(N / (TPB * PPT));  // 4096 blocks
  Transformer_warp_kernel<<<grid, TPB, 0, stream>>>(flow, pqf, out);
}